// LSTM_64106681860366
// MI455X (gfx1250) — compile-verified
//
#include <hip/hip_runtime.h>
#include <math.h>
#include <stdint.h>

// LSTM cell, H = I = 4096, fp32.
// Phase 1: z[4H] = Wcat @ x + Whcat @ h via V_WMMA_F32_16X16X4_F32 with
//   double-buffered GLOBAL_LOAD_ASYNC_TO_LDS_B128 tile staging (ASYNCcnt).
//   Each wave owns 16 output rows and a private 16x32 LDS tile pair; no
//   block barriers needed. Only column N=0 of D is consumed (lanes 0/16).
// Phase 2: pointwise gates -> h_t, c_t; out = [h_t, h_t, c_t].

typedef __attribute__((ext_vector_type(2))) float v2f;
typedef __attribute__((ext_vector_type(8))) float v8f;

#define HSZ 4096
#define ISZ 4096
#define KC 32        // K tile (floats)
#define PAD 36       // LDS row stride in floats (conflict-free b64 reads)
#define WAVES 8      // waves per block
#define NROWBLK 1024 // 16384 rows / 16
#define NT 256       // 128 K-tiles per matrix x 2 matrices

__global__ __launch_bounds__(256) void lstm_gemv_wmma(
    const float* __restrict__ x, const float* __restrict__ h,
    const float* __restrict__ W_ii, const float* __restrict__ W_if,
    const float* __restrict__ W_ig, const float* __restrict__ W_io,
    const float* __restrict__ W_hi, const float* __restrict__ W_hf,
    const float* __restrict__ W_hg, const float* __restrict__ W_ho,
    float* __restrict__ z)
{
    __shared__ float lds[2][WAVES][16][PAD];   // 36 KB

    const int tid  = threadIdx.x;
    const int lane = tid & 31;
    const int w    = tid >> 5;
    const int rb   = blockIdx.x * WAVES + w;   // 0..1023 row-blocks of 16
    const int gate = rb >> 8;                  // 256 row-blocks per gate
    const int r0   = (rb & 255) << 4;          // first row inside gate matrix

    const float* Wi = (gate == 0) ? W_ii : (gate == 1) ? W_if
                    : (gate == 2) ? W_ig : W_io;
    const float* Wh = (gate == 0) ? W_hi : (gate == 1) ? W_hf
                    : (gate == 2) ? W_hg : W_ho;

    const int m    = lane & 15;   // M index of this lane's A fragment
    const int half = lane >> 4;   // selects K pair {0,1} vs {2,3}

    // Per-lane staging coordinates: 16 rows x 8 float4 chunks = 128 chunks,
    // 4 async b128 instructions x 32 lanes.  q = i*32+lane -> row=q>>3,
    // c4=(q&7)*4: each row is covered by 8 lanes = 128B contiguous.
    const int srow[4] = { (0 * 32 + lane) >> 3, (1 * 32 + lane) >> 3,
                          (2 * 32 + lane) >> 3, (3 * 32 + lane) >> 3 };
    const int sc4     = (lane & 7) << 2;

    // Issue one tile's worth of async global->LDS b128 DMA (4 instructions).
    auto stage = [&](int t, int buf) {
        const float* __restrict__ W = (t < NT / 2) ? Wi : Wh;
        const int k0 = (t & (NT / 2 - 1)) * KC;
        #pragma unroll
        for (int i = 0; i < 4; ++i) {
            const int row = srow[i];
            const float* gp = W + (size_t)(r0 + row) * ISZ + k0 + sc4;
            const uint32_t loff =
                (uint32_t)(uintptr_t)(void*)&lds[buf][w][row][sc4];
            asm volatile("global_load_async_to_lds_b128 %0, %1, off"
                         :: "v"(loff), "v"((uint64_t)(uintptr_t)gp)
                         : "memory");
        }
    };

    v8f acc = {};                 // 16x16 f32 accumulator, col 0 is the GEMV

    stage(0, 0);
    for (int t = 0; t < NT; ++t) {
        const int buf = t & 1;
        if (t + 1 < NT) {
            stage(t + 1, buf ^ 1);          // prefetch next tile
            asm volatile("s_wait_asynccnt 0x4" ::: "memory");  // tile t done
        } else {
            asm volatile("s_wait_asynccnt 0x0" ::: "memory");
        }

        const float* __restrict__ v = (t < NT / 2) ? x : h;
        const int k0 = (t & (NT / 2 - 1)) * KC;

        #pragma unroll
        for (int kk = 0; kk < 8; ++kk) {
            const int kb = (kk << 2) + (half << 1);
            // A fragment: lane holds W[r0+m, k0+kb], W[r0+m, k0+kb+1]
            v2f a = *reinterpret_cast<const v2f*>(&lds[buf][w][m][kb]);
            // B fragment: vector chunk broadcast to every column
            // (only column N=0 of D is read; other columns are don't-care)
            v2f b = *reinterpret_cast<const v2f*>(v + k0 + kb);
            acc = __builtin_amdgcn_wmma_f32_16x16x4_f32(
                    false, a, false, b, (short)0, acc, false, false);
        }
    }

    // ---- write column N=0: lane 0 -> M=0..7, lane 16 -> M=8..15 ----
    const int zr = rb << 4;       // global row in z = gate*H + r0
    if (lane == 0) {
        #pragma unroll
        for (int j = 0; j < 8; ++j) z[zr + j] = acc[j];
    } else if (lane == 16) {
        #pragma unroll
        for (int j = 0; j < 8; ++j) z[zr + 8 + j] = acc[j];
    }
}

__global__ __launch_bounds__(256) void lstm_pointwise(
    const float* __restrict__ z, const float* __restrict__ c_prev,
    const float* __restrict__ b_ii, const float* __restrict__ b_if,
    const float* __restrict__ b_ig, const float* __restrict__ b_io,
    const float* __restrict__ b_hi, const float* __restrict__ b_hf,
    const float* __restrict__ b_hg, const float* __restrict__ b_ho,
    float* __restrict__ out)
{
    const int r = blockIdx.x * blockDim.x + threadIdx.x;
    if (r >= HSZ) return;

    const float zi = z[0 * HSZ + r] + b_ii[r] + b_hi[r];
    const float zf = z[1 * HSZ + r] + b_if[r] + b_hf[r];
    const float zg = z[2 * HSZ + r] + b_ig[r] + b_hg[r];
    const float zo = z[3 * HSZ + r] + b_io[r] + b_ho[r];

    const float it = 1.0f / (1.0f + expf(-zi));
    const float ft = 1.0f / (1.0f + expf(-zf));
    const float gt = tanhf(zg);
    const float ot = 1.0f / (1.0f + expf(-zo));

    const float ct = ft * c_prev[r] + it * gt;
    const float ht = ot * tanhf(ct);

    out[0 * HSZ + r] = ht;   // h_t
    out[1 * HSZ + r] = ht;   // h_t (tuple repeat)
    out[2 * HSZ + r] = ct;   // c_t
}

extern "C" void kernel_launch(void* const* d_in, const int* in_sizes, int n_in,
                              void* d_out, int out_size, void* d_ws, size_t ws_size,
                              hipStream_t stream) {
    const float* x      = (const float*)d_in[0];
    const float* h_prev = (const float*)d_in[1];
    const float* c_prev = (const float*)d_in[2];
    const float* W_ii   = (const float*)d_in[3];
    const float* W_if   = (const float*)d_in[4];
    const float* W_ig   = (const float*)d_in[5];
    const float* W_io   = (const float*)d_in[6];
    const float* W_hi   = (const float*)d_in[7];
    const float* W_hf   = (const float*)d_in[8];
    const float* W_hg   = (const float*)d_in[9];
    const float* W_ho   = (const float*)d_in[10];
    const float* b_ii   = (const float*)d_in[11];
    const float* b_if   = (const float*)d_in[12];
    const float* b_ig   = (const float*)d_in[13];
    const float* b_io   = (const float*)d_in[14];
    const float* b_hi   = (const float*)d_in[15];
    const float* b_hf   = (const float*)d_in[16];
    const float* b_hg   = (const float*)d_in[17];
    const float* b_ho   = (const float*)d_in[18];

    float* zbuf = (float*)d_ws;          // 4*H floats = 64 KB scratch
    float* out  = (float*)d_out;

    lstm_gemv_wmma<<<NROWBLK / WAVES, 256, 0, stream>>>(
        x, h_prev, W_ii, W_if, W_ig, W_io, W_hi, W_hf, W_hg, W_ho, zbuf);

    lstm_pointwise<<<(HSZ + 255) / 256, 256, 0, stream>>>(
        zbuf, c_prev, b_ii, b_if, b_ig, b_io, b_hi, b_hf, b_hg, b_ho, out);
}